// GraphAttention_45724221834028
// MI455X (gfx1250) — compile-verified
//
#include <hip/hip_runtime.h>
#include <hip/hip_bf16.h>
#include <stdint.h>

#define D_FEAT 128
#define UNITS 128

typedef float v2f __attribute__((ext_vector_type(2)));
typedef float v8f __attribute__((ext_vector_type(8)));

// ---------------------------------------------------------------------------
// Kernel 0: zero output + sum_e accumulators (graph-capture safe, no memset)
// ---------------------------------------------------------------------------
__global__ void gat_zero_kernel(float* __restrict__ out, float* __restrict__ sum_e,
                                int n_out, int n_sum) {
    int stride = gridDim.x * blockDim.x;
    for (int i = blockIdx.x * blockDim.x + threadIdx.x; i < n_out; i += stride)
        out[i] = 0.0f;
    for (int i = blockIdx.x * blockDim.x + threadIdx.x; i < n_sum; i += stride)
        sum_e[i] = 0.0f;
}

// ---------------------------------------------------------------------------
// Kernel 1: H = X @ W  via V_WMMA_F32_16X16X4_F32 (fp32, reference precision)
// Block = 256 threads = 8 waves. Block computes a 16(M) x 128(N) tile of H.
// Wave w owns the 16x16 tile at columns [16w, 16w+16). K=128 -> 32 WMMA steps.
// A-tile (16x128 fp32 = 8 KB) staged in LDS once per block, shared by 8 waves.
// ---------------------------------------------------------------------------
__global__ __launch_bounds__(256) void gat_gemm_wmma_kernel(
    const float* __restrict__ X,   // [n_nodes, 128]
    const float* __restrict__ W,   // [128, 128]
    float* __restrict__ H,         // [n_nodes, 128]
    int n_nodes) {
    const int wave = threadIdx.x >> 5;   // 0..7 -> output column tile
    const int lane = threadIdx.x & 31;
    const int m0 = blockIdx.x * 16;
    const int n0 = wave * 16;

    __shared__ float As[16][D_FEAT + 4];   // +4 pad to dodge bank conflicts

    // cooperative load of A tile (2048 floats / 256 threads = 8 each)
    for (int i = threadIdx.x; i < 16 * D_FEAT; i += 256) {
        int r = i >> 7;            // /128
        int c = i & (D_FEAT - 1);
        int gr = m0 + r;
        As[r][c] = (gr < n_nodes) ? X[(size_t)gr * D_FEAT + c] : 0.0f;
    }
    __syncthreads();

    // ISA 7.12.2 "32-bit A-Matrix 16x4": lanes 0-15 -> M=lane, v0=K0,v1=K1;
    // lanes 16-31 -> M=lane-16, v0=K2,v1=K3.  B (4x16) mirrored on K.
    const int mA = lane & 15;
    const int kh = (lane >> 4) << 1;       // 0 or 2
    const int nB = lane & 15;

    v8f acc = {};
#pragma unroll
    for (int kk = 0; kk < D_FEAT; kk += 4) {
        v2f a, b;
        a.x = As[mA][kk + kh];
        a.y = As[mA][kk + kh + 1];
        b.x = W[(size_t)(kk + kh) * UNITS + n0 + nB];
        b.y = W[(size_t)(kk + kh + 1) * UNITS + n0 + nB];
        acc = __builtin_amdgcn_wmma_f32_16x16x4_f32(
            /*neg_a=*/false, a, /*neg_b=*/false, b,
            /*c_mod=*/(short)0, acc, /*reuse_a=*/false, /*reuse_b=*/false);
    }

    // D layout: VGPR r -> M = r (lanes 0-15) or r+8 (lanes 16-31), N = lane&15
    const int colD = n0 + (lane & 15);
    const int rowBase = m0 + ((lane >> 4) << 3);
#pragma unroll
    for (int r = 0; r < 8; ++r) {
        int row = rowBase + r;
        if (row < n_nodes) H[(size_t)row * UNITS + colD] = acc[r];
    }
}

// ---------------------------------------------------------------------------
// Kernel 2: per-node attention scores. One wave32 per node, lane = 4 features.
// ---------------------------------------------------------------------------
__global__ __launch_bounds__(256) void gat_score_kernel(
    const float* __restrict__ H,
    const float* __restrict__ KA,      // [256]: ka_t = [0:128), ka_s = [128:256)
    float* __restrict__ score_t,
    float* __restrict__ score_s,
    int n_nodes) {
    int node = blockIdx.x * 8 + (threadIdx.x >> 5);
    int lane = threadIdx.x & 31;
    if (node >= n_nodes) return;

    const float4* hp = (const float4*)(H + (size_t)node * UNITS);
    const float4* kt = (const float4*)(KA);
    const float4* ks = (const float4*)(KA + UNITS);
    float4 h = hp[lane];
    float4 t = kt[lane];
    float4 s = ks[lane];
    float st = h.x * t.x + h.y * t.y + h.z * t.z + h.w * t.w;
    float ss = h.x * s.x + h.y * s.y + h.z * s.z + h.w * s.w;
#pragma unroll
    for (int off = 16; off > 0; off >>= 1) {
        st += __shfl_xor(st, off, 32);
        ss += __shfl_xor(ss, off, 32);
    }
    if (lane == 0) {
        score_t[node] = st;
        score_s[node] = ss;
    }
}

// ---------------------------------------------------------------------------
// Kernel 3: per-edge logits: leaky-relu, clip, exp; accumulate softmax denom.
// ---------------------------------------------------------------------------
__global__ void gat_edge_score_kernel(
    const long long* __restrict__ edges,   // [n_edges, 2] int64 (tgt, src)
    const float* __restrict__ score_t,
    const float* __restrict__ score_s,
    float* __restrict__ e_val,
    float* __restrict__ sum_e,
    int n_edges) {
    int e = blockIdx.x * blockDim.x + threadIdx.x;
    if (e >= n_edges) return;
    int tgt = (int)edges[2 * (size_t)e + 0];
    int src = (int)edges[2 * (size_t)e + 1];
    float v = score_t[tgt] + score_s[src];
    v = (v > 0.0f) ? v : 0.2f * v;               // leaky_relu, alpha=0.2
    v = fminf(fmaxf(v, -2.0f), 2.0f);            // clip
    v = __expf(v);
    e_val[e] = v;
    atomicAdd(&sum_e[tgt], v);
}

// ---------------------------------------------------------------------------
// Kernel 4: weighted message aggregation. One wave32 per edge; lane owns 4
// features. h[] and out[] are L2-resident (51 MB each << 192 MB L2), so the
// gather + global_atomic_add_f32 scatter stays on-chip.
// ---------------------------------------------------------------------------
__global__ __launch_bounds__(256) void gat_edge_agg_kernel(
    const long long* __restrict__ edges,
    const float* __restrict__ H,
    const float* __restrict__ e_val,
    const float* __restrict__ sum_e,
    float* __restrict__ out,
    int n_edges) {
    int e = blockIdx.x * 8 + (threadIdx.x >> 5);
    int lane = threadIdx.x & 31;
    if (e >= n_edges) return;
    int tgt = (int)edges[2 * (size_t)e + 0];
    int src = (int)edges[2 * (size_t)e + 1];
    float att = e_val[e] / (sum_e[tgt] + 1e-9f);
    const float4* hs = (const float4*)(H + (size_t)src * UNITS);
    float4 m = hs[lane];
    float* op = out + (size_t)tgt * UNITS + lane * 4;
    atomicAdd(op + 0, m.x * att);
    atomicAdd(op + 1, m.y * att);
    atomicAdd(op + 2, m.z * att);
    atomicAdd(op + 3, m.w * att);
}

// ---------------------------------------------------------------------------
// Host-side launch
// ---------------------------------------------------------------------------
static inline size_t align_up(size_t x, size_t a) { return (x + a - 1) & ~(a - 1); }

extern "C" void kernel_launch(void* const* d_in, const int* in_sizes, int n_in,
                              void* d_out, int out_size, void* d_ws, size_t ws_size,
                              hipStream_t stream) {
    const float* X        = (const float*)d_in[0];       // node_states [N,128]
    const long long* edges = (const long long*)d_in[1];  // [E,2] int64
    const float* W        = (const float*)d_in[2];       // kernel [128,128]
    const float* KA       = (const float*)d_in[3];       // kernel_attention [256]
    float* out            = (float*)d_out;

    const int n_nodes = in_sizes[0] / D_FEAT;
    const int n_edges = in_sizes[1] / 2;

    // workspace layout
    char* ws = (char*)d_ws;
    size_t off = 0;
    float* H = (float*)(ws + off);        off = align_up(off + (size_t)n_nodes * UNITS * sizeof(float), 256);
    float* score_t = (float*)(ws + off);  off = align_up(off + (size_t)n_nodes * sizeof(float), 256);
    float* score_s = (float*)(ws + off);  off = align_up(off + (size_t)n_nodes * sizeof(float), 256);
    float* e_val = (float*)(ws + off);    off = align_up(off + (size_t)n_edges * sizeof(float), 256);
    float* sum_e = (float*)(ws + off);    off = align_up(off + (size_t)n_nodes * sizeof(float), 256);
    (void)ws_size;

    // 0) zero accumulators (out and sum_e are scatter-add targets)
    gat_zero_kernel<<<2048, 256, 0, stream>>>(out, sum_e, n_nodes * UNITS, n_nodes);

    // 1) H = X @ W (fp32 WMMA)
    int gemm_blocks = (n_nodes + 15) / 16;
    gat_gemm_wmma_kernel<<<gemm_blocks, 256, 0, stream>>>(X, W, H, n_nodes);

    // 2) scores
    int score_blocks = (n_nodes + 7) / 8;
    gat_score_kernel<<<score_blocks, 256, 0, stream>>>(H, KA, score_t, score_s, n_nodes);

    // 3) edge logits + denom
    int es_blocks = (n_edges + 255) / 256;
    gat_edge_score_kernel<<<es_blocks, 256, 0, stream>>>(edges, score_t, score_s,
                                                         e_val, sum_e, n_edges);

    // 4) aggregation
    int agg_blocks = (n_edges + 7) / 8;
    gat_edge_agg_kernel<<<agg_blocks, 256, 0, stream>>>(edges, H, e_val, sum_e,
                                                        out, n_edges);
}